// GraphConvolution1_81887846466078
// MI455X (gfx1250) — compile-verified
//
#include <hip/hip_runtime.h>

typedef __attribute__((ext_vector_type(2))) float v2f;
typedef __attribute__((ext_vector_type(8))) float v8f;

#define D_IN  128
#define D_OUT 64

// ---------------------------------------------------------------------------
// Kernel 1: h = x @ W via V_WMMA_F32_16X16X4_F32.
// Block = 256 threads = 8 waves. Each wave computes one 16-row M tile across
// all 4 N tiles (D_OUT = 64). W is staged once per block into LDS, already
// permuted into the WMMA B-matrix register layout so the hot loop issues a
// single conflict-free ds_load_b64 per B fragment.
// ---------------------------------------------------------------------------
__global__ __launch_bounds__(256) void gemm_h_kernel(
    const float* __restrict__ X, const float* __restrict__ W,
    float* __restrict__ H, int nTilesM) {
  // sB[kc][nt][lane] : float2 = the two f32 values lane needs for B chunk kc,
  // N-tile nt.  32 kc * 4 nt * 32 lanes * 8B = 32 KB.
  __shared__ float2 sB[32 * 4 * 32];

  const int tid = threadIdx.x;

  // Stage W (128x64 row-major) into WMMA B layout.
  // B is 4x16 (KxN): lanes 0-15 hold K={0,1}, lanes 16-31 hold K={2,3}.
  #pragma unroll
  for (int i = 0; i < 16; ++i) {
    const int idx   = tid + i * 256;          // 0..4095
    const int lane  = idx & 31;
    const int nt    = (idx >> 5) & 3;
    const int kc    = idx >> 7;
    const int kOff  = (lane >> 4) * 2;
    const int laneN = lane & 15;
    const int row0  = kc * 4 + kOff;          // K index of first element
    float2 v;
    v.x = W[row0 * D_OUT + nt * 16 + laneN];
    v.y = W[(row0 + 1) * D_OUT + nt * 16 + laneN];
    sB[idx] = v;
  }
  __syncthreads();

  const int wave  = tid >> 5;
  const int lane  = tid & 31;
  const int tileM = blockIdx.x * 8 + wave;
  if (tileM >= nTilesM) return;               // wave-uniform: EXEC stays all-1s

  // A is 16x4 (MxK): lanes 0-15 hold M=lane, K={0,1}; lanes 16-31 K={2,3}.
  const int    kOff = (lane >> 4) * 2;
  const int    rowA = tileM * 16 + (lane & 15);
  const float* xrow = X + (size_t)rowA * D_IN + kOff;

  v8f c0 = {}, c1 = {}, c2 = {}, c3 = {};

  #pragma unroll 4
  for (int kc = 0; kc < 32; ++kc) {
    v2f a = *(const v2f*)(xrow + kc * 4);
    const float2* bBase = &sB[kc * 128 + lane];
    v2f b0 = *(const v2f*)(bBase + 0 * 32);
    v2f b1 = *(const v2f*)(bBase + 1 * 32);
    v2f b2 = *(const v2f*)(bBase + 2 * 32);
    v2f b3 = *(const v2f*)(bBase + 3 * 32);
    c0 = __builtin_amdgcn_wmma_f32_16x16x4_f32(false, a, false, b0, (short)0, c0, false, false);
    c1 = __builtin_amdgcn_wmma_f32_16x16x4_f32(false, a, false, b1, (short)0, c1, false, false);
    c2 = __builtin_amdgcn_wmma_f32_16x16x4_f32(false, a, false, b2, (short)0, c2, false, false);
    c3 = __builtin_amdgcn_wmma_f32_16x16x4_f32(false, a, false, b3, (short)0, c3, false, false);
  }

  // C/D layout: VGPR r holds M=r (lanes 0-15) and M=r+8 (lanes 16-31), N=lane&15.
  float* hBase = H + (size_t)(tileM * 16 + ((lane >> 4) << 3)) * D_OUT + (lane & 15);
  #pragma unroll
  for (int r = 0; r < 8; ++r) {
    hBase[(size_t)r * D_OUT +  0] = c0[r];
    hBase[(size_t)r * D_OUT + 16] = c1[r];
    hBase[(size_t)r * D_OUT + 32] = c2[r];
    hBase[(size_t)r * D_OUT + 48] = c3[r];
  }
}

// ---------------------------------------------------------------------------
// Kernel 2: zero the accumulation buffer (d_out), float4-wide.
// ---------------------------------------------------------------------------
__global__ __launch_bounds__(256) void zero_out_kernel(float4* __restrict__ out, int n4) {
  int i = blockIdx.x * blockDim.x + threadIdx.x;
  if (i < n4) out[i] = make_float4(0.f, 0.f, 0.f, 0.f);
}

// ---------------------------------------------------------------------------
// Kernel 3: scatter-add  out[dst] += h[src]  over all edges.
// One wave owns 32 edges: coalesced index loads, then v_readlane_b32 to
// scalarize (s, d); each lane covers 2 of the 64 dims (b64 load + 2 atomics).
// h and out are L2-resident (25.6 MB each, 192 MB L2).
// ---------------------------------------------------------------------------
__global__ __launch_bounds__(256) void scatter_edges_kernel(
    const float* __restrict__ H, const int* __restrict__ srcI,
    const int* __restrict__ dstI, float* __restrict__ out, int nEdges) {
  const int lane = threadIdx.x & 31;
  const int wid  = (blockIdx.x * blockDim.x + threadIdx.x) >> 5;
  const int base = wid * 32;
  if (base >= nEdges) return;

  int myS = 0, myD = 0;
  if (base + lane < nEdges) {
    myS = srcI[base + lane];
    myD = dstI[base + lane];
  }
  const int valid = (nEdges - base < 32) ? (nEdges - base) : 32;

  for (int j = 0; j < valid; ++j) {
    const int s = __builtin_amdgcn_readlane(myS, j);   // SGPR address base
    const int d = __builtin_amdgcn_readlane(myD, j);
    const float2 hv = *(const float2*)(H + (size_t)s * D_OUT + lane * 2);
    float* o = out + (size_t)d * D_OUT + lane * 2;
    atomicAdd(o,     hv.x);
    atomicAdd(o + 1, hv.y);
  }
}

// ---------------------------------------------------------------------------
// Kernel 4: out = relu(out + b), float4-wide in place.
// ---------------------------------------------------------------------------
__global__ __launch_bounds__(256) void bias_relu_kernel(
    float4* __restrict__ out, const float* __restrict__ b, int n4) {
  int i = blockIdx.x * blockDim.x + threadIdx.x;
  if (i < n4) {
    float4 v = out[i];
    const int c = (i * 4) & (D_OUT - 1);
    v.x = fmaxf(v.x + b[c + 0], 0.f);
    v.y = fmaxf(v.y + b[c + 1], 0.f);
    v.z = fmaxf(v.z + b[c + 2], 0.f);
    v.w = fmaxf(v.w + b[c + 3], 0.f);
    out[i] = v;
  }
}

extern "C" void kernel_launch(void* const* d_in, const int* in_sizes, int n_in,
                              void* d_out, int out_size, void* d_ws, size_t ws_size,
                              hipStream_t stream) {
  const float* X  = (const float*)d_in[0];           // [N, 128]
  const int*   EI = (const int*)d_in[1];             // [2, E]
  const float* W  = (const float*)d_in[2];           // [128, 64]
  const float* B  = (const float*)d_in[3];           // [64]
  float*       out = (float*)d_out;                  // [N, 64]
  float*       H   = (float*)d_ws;                   // [N, 64] scratch

  const int nNodes = in_sizes[0] / D_IN;             // 100000 (multiple of 16)
  const int nEdges = in_sizes[1] / 2;                // 1600000
  const int* srcI = EI;
  const int* dstI = EI + nEdges;

  // 1) h = x @ W  (WMMA)
  const int nTilesM = nNodes / 16;
  gemm_h_kernel<<<(nTilesM + 7) / 8, 256, 0, stream>>>(X, W, H, nTilesM);

  // 2) out = 0
  const int n  = nNodes * D_OUT;
  const int n4 = n / 4;
  zero_out_kernel<<<(n4 + 255) / 256, 256, 0, stream>>>((float4*)out, n4);

  // 3) out[dst] += h[src]
  const int nWaves = (nEdges + 31) / 32;
  scatter_edges_kernel<<<(nWaves + 7) / 8, 256, 0, stream>>>(H, srcI, dstI, out, nEdges);

  // 4) out = relu(out + b)
  bias_relu_kernel<<<(n4 + 255) / 256, 256, 0, stream>>>((float4*)out, B, n4);
}